// RPN_3D_loss_16209206575853
// MI455X (gfx1250) — compile-verified
//
#include <hip/hip_runtime.h>

#define H_FEAT 32
#define W_FEAT 110
#define STRIDE_PX 16
#define NUM_ANCH 36
#define NUM_CLS 4
#define MAX_GTS 32
#define BATCH 4
#define A_TOTAL (H_FEAT * W_FEAT * NUM_ANCH)   /* 126720 */
#define BLK 256
#define BLOCKS_PER_B (A_TOTAL / BLK)           /* 495 */
#define NPART (BLOCKS_PER_B * BATCH)           /* 1980 */

typedef float v2f __attribute__((ext_vector_type(2)));
typedef float v8f __attribute__((ext_vector_type(8)));

__device__ __forceinline__ float smooth_l1(float d) {
    float ad = fabsf(d);
    return ad < 1.0f ? 0.5f * d * d : ad - 0.5f;
}

// ---------------------------------------------------------------------------
// Kernel 0: zero the best-per-gt table (int-encoded non-negative floats).
// ---------------------------------------------------------------------------
__global__ void k_init(int* __restrict__ g_best) {
    int t = threadIdx.x;
    if (t < BATCH * MAX_GTS) g_best[t] = 0;   // bit pattern of +0.0f
}

// ---------------------------------------------------------------------------
// Kernel 1: best_per_gt[b][g] = max over all anchors of IoU (valid gts only).
// Integer atomicMax on non-negative float bit patterns is order-independent,
// hence deterministic.
// ---------------------------------------------------------------------------
__global__ void k_best(const float* __restrict__ anchors,
                       const float* __restrict__ gt2d,
                       const int*   __restrict__ glab,
                       int*         __restrict__ g_best) {
    __shared__ float s_gt[MAX_GTS * 4];
    __shared__ int   s_val[MAX_GTS];
    __shared__ int   s_bmax[MAX_GTS];
    __shared__ float s_anch[NUM_ANCH * 4];

    const int b = blockIdx.y;
    const int tid = threadIdx.x;
    if (tid < MAX_GTS * 4) s_gt[tid] = gt2d[b * MAX_GTS * 4 + tid];
    if (tid < MAX_GTS) { s_val[tid] = (glab[b * MAX_GTS + tid] > 0); s_bmax[tid] = 0; }
    if (tid < NUM_ANCH * 4) { int t = tid >> 2, c = tid & 3; s_anch[tid] = anchors[t * 9 + c]; }
    __syncthreads();

    const int a    = blockIdx.x * BLK + tid;
    const int cell = a / NUM_ANCH;
    const int t    = a % NUM_ANCH;
    const float sx = (float)((cell % W_FEAT) * STRIDE_PX);
    const float sy = (float)((cell / W_FEAT) * STRIDE_PX);
    const float r0 = sx + s_anch[t * 4 + 0];
    const float r1 = sy + s_anch[t * 4 + 1];
    const float r2 = sx + s_anch[t * 4 + 2];
    const float r3 = sy + s_anch[t * 4 + 3];
    const float ar = (r2 - r0 + 1.0f) * (r3 - r1 + 1.0f);

    const int lane = tid & 31;
    for (int g = 0; g < MAX_GTS; ++g) {
        if (!s_val[g]) continue;               // uniform across the block
        const float q0 = s_gt[4 * g], q1 = s_gt[4 * g + 1];
        const float q2 = s_gt[4 * g + 2], q3 = s_gt[4 * g + 3];
        float ix = fminf(r2, q2) - fmaxf(r0, q0) + 1.0f; ix = fmaxf(ix, 0.0f);
        float iy = fminf(r3, q3) - fmaxf(r1, q1) + 1.0f; iy = fmaxf(iy, 0.0f);
        const float inter = ix * iy;
        const float ag    = (q2 - q0 + 1.0f) * (q3 - q1 + 1.0f);
        float iou = inter / (ar + ag - inter);  // >= 0
        for (int off = 16; off; off >>= 1) iou = fmaxf(iou, __shfl_down(iou, off));
        if (lane == 0) atomicMax(&s_bmax[g], __float_as_int(iou));
    }
    __syncthreads();
    if (tid < MAX_GTS) atomicMax(&g_best[b * MAX_GTS + tid], s_bmax[tid]);
}

// ---------------------------------------------------------------------------
// Kernel 2: per-anchor matching, CE, regression targets, smooth-L1; block
// reduction to 5 partial sums per block (deterministic, no float atomics).
// ---------------------------------------------------------------------------
__global__ void k_main(const float4* __restrict__ cls,
                       const float4* __restrict__ bbox2d,
                       const float*  __restrict__ bbox3d,
                       const float*  __restrict__ anchors,
                       const float*  __restrict__ means,
                       const float*  __restrict__ stds,
                       const float*  __restrict__ gt2d,
                       const float*  __restrict__ gt3d,
                       const int*    __restrict__ glabels,
                       const int*    __restrict__ g_best,
                       float*        __restrict__ partials) {
    __shared__ float s_gt2[MAX_GTS * 4];
    __shared__ float s_gt3[MAX_GTS * 7];
    __shared__ int   s_lab[MAX_GTS];
    __shared__ float s_best[MAX_GTS];
    __shared__ float s_anch[NUM_ANCH * 9];
    __shared__ float s_mean[11], s_std[11];
    __shared__ float s_red[8][5];

    const int b = blockIdx.y;
    const int tid = threadIdx.x;
    if (tid < MAX_GTS * 4) s_gt2[tid] = gt2d[b * MAX_GTS * 4 + tid];
    if (tid < MAX_GTS * 7) s_gt3[tid] = gt3d[b * MAX_GTS * 7 + tid];
    if (tid < MAX_GTS) {
        s_lab[tid]  = glabels[b * MAX_GTS + tid];
        s_best[tid] = __int_as_float(g_best[b * MAX_GTS + tid]);
    }
    for (int i = tid; i < NUM_ANCH * 9; i += BLK) s_anch[i] = anchors[i];
    if (tid < 11) { s_mean[tid] = means[tid]; s_std[tid] = stds[tid]; }
    __syncthreads();

    const int a    = blockIdx.x * BLK + tid;
    const int cell = a / NUM_ANCH;
    const int t    = a % NUM_ANCH;
    const float* src = &s_anch[t * 9];
    const float sx = (float)((cell % W_FEAT) * STRIDE_PX);
    const float sy = (float)((cell / W_FEAT) * STRIDE_PX);
    const float r0 = sx + src[0], r1 = sy + src[1];
    const float r2 = sx + src[2], r3 = sy + src[3];
    const float wid = r2 - r0 + 1.0f, hei = r3 - r1 + 1.0f;
    const float cx  = r0 + 0.5f * wid, cy = r1 + 0.5f * hei;
    const float ar  = wid * hei;

    float maxiou = -INFINITY;
    int   arg    = 0;
    bool  anyb   = false;
    for (int g = 0; g < MAX_GTS; ++g) {
        const bool valid = s_lab[g] > 0;
        const float q0 = s_gt2[4 * g], q1 = s_gt2[4 * g + 1];
        const float q2 = s_gt2[4 * g + 2], q3 = s_gt2[4 * g + 3];
        float ix = fminf(r2, q2) - fmaxf(r0, q0) + 1.0f; ix = fmaxf(ix, 0.0f);
        float iy = fminf(r3, q3) - fmaxf(r1, q1) + 1.0f; iy = fmaxf(iy, 0.0f);
        const float inter = ix * iy;
        const float ag    = (q2 - q0 + 1.0f) * (q3 - q1 + 1.0f);
        const float iou   = inter / (ar + ag - inter);
        const float iou_m = valid ? iou : -1.0f;
        if (iou_m > maxiou) { maxiou = iou_m; arg = g; }   // first-max wins, as jnp.argmax
        if (valid && iou_m >= s_best[g] && s_best[g] >= 0.35f) anyb = true;
    }
    const bool fg = (maxiou >= 0.5f) || anyb;
    const bool bg = !fg && (maxiou < 0.5f) && (maxiou >= 0.0f);

    // classification loss (log-softmax over 4 classes)
    const float4 lg = cls[(size_t)b * A_TOTAL + a];
    const int label = fg ? s_lab[arg] : 0;
    const float mx  = fmaxf(fmaxf(lg.x, lg.y), fmaxf(lg.z, lg.w));
    const float lse = mx + logf(expf(lg.x - mx) + expf(lg.y - mx) +
                                expf(lg.z - mx) + expf(lg.w - mx));
    const float xl  = (label == 0) ? lg.x : (label == 1) ? lg.y : (label == 2) ? lg.z : lg.w;
    const float ce  = lse - xl;
    const float wgt = (fg || bg) ? 1.0f : 0.0f;

    float s0 = ce * wgt, s1 = wgt, s2 = fg ? 1.0f : 0.0f, s3 = 0.0f, s4 = 0.0f;
    if (fg) {
        const float* G2 = &s_gt2[4 * arg];
        const float* G3 = &s_gt3[7 * arg];
        const float gw  = G2[2] - G2[0] + 1.0f, gh = G2[3] - G2[1] + 1.0f;
        const float gcx = G2[0] + 0.5f * gw,    gcy = G2[1] + 0.5f * gh;
        float t2[4] = { (gcx - cx) / wid, (gcy - cy) / hei, logf(gw / wid), logf(gh / hei) };
        const float4 p2 = bbox2d[(size_t)b * A_TOTAL + a];
        const float pv[4] = { p2.x, p2.y, p2.z, p2.w };
        #pragma unroll
        for (int j = 0; j < 4; ++j)
            s3 += smooth_l1(pv[j] - (t2[j] - s_mean[j]) / s_std[j]);

        float t3[7] = { (G3[0] - cx) / wid, (G3[1] - cy) / hei, G3[2] - src[4],
                        logf(G3[3] / src[5]), logf(G3[4] / src[6]),
                        logf(G3[5] / src[7]), G3[6] - src[8] };
        const float* p3 = bbox3d + ((size_t)b * A_TOTAL + a) * 7;
        #pragma unroll
        for (int j = 0; j < 7; ++j)
            s4 += smooth_l1(p3[j] - (t3[j] - s_mean[4 + j]) / s_std[4 + j]);
    }

    // deterministic block reduction of the 5 sums
    float v[5] = { s0, s1, s2, s3, s4 };
    const int lane = tid & 31, wv = tid >> 5;
    #pragma unroll
    for (int q = 0; q < 5; ++q)
        for (int off = 16; off; off >>= 1) v[q] += __shfl_down(v[q], off);
    if (lane == 0) {
        #pragma unroll
        for (int q = 0; q < 5; ++q) s_red[wv][q] = v[q];
    }
    __syncthreads();
    if (tid == 0) {
        const int blk = blockIdx.y * gridDim.x + blockIdx.x;
        #pragma unroll
        for (int q = 0; q < 5; ++q) {
            float s = 0.0f;
            for (int i = 0; i < 8; ++i) s += s_red[i][q];
            partials[blk * 5 + q] = s;
        }
    }
}

// ---------------------------------------------------------------------------
// Kernel 3: single-wave final reduction. Cross-lane sum is done on the matrix
// unit: D = A x ones + 0 with V_WMMA_F32_16X16X4_F32. Per the ISA 16x4 f32
// A-layout, lane m (VGPR0) and lane m+16 (VGPR0, K=2) feed row m, so
// D[m,*] = s[m] + s[m+16]; summing the 8 D VGPRs in lanes 0 and 16 gives the
// full 32-lane sum.
// ---------------------------------------------------------------------------
__global__ void k_final(const float* __restrict__ partials, float* __restrict__ out) {
    const int lane = threadIdx.x;  // 32 threads = 1 wave, EXEC all ones at WMMA
    float s[5] = { 0.0f, 0.0f, 0.0f, 0.0f, 0.0f };
    for (int i = lane; i < NPART; i += 32) {
        const float* p = partials + i * 5;
        s[0] += p[0]; s[1] += p[1]; s[2] += p[2]; s[3] += p[3]; s[4] += p[4];
    }

    v2f ones; ones.x = 1.0f; ones.y = 1.0f;
    float tot[5];
    #pragma unroll
    for (int q = 0; q < 5; ++q) {
        v2f av; av.x = s[q]; av.y = 0.0f;
        v8f c = {};
        // 8 args: (neg_a, A, neg_b, B, c_mod, C, reuse_a, reuse_b)
        v8f d = __builtin_amdgcn_wmma_f32_16x16x4_f32(
            false, av, false, ones, (short)0, c, false, false);
        float tl = d[0] + d[1] + d[2] + d[3] + d[4] + d[5] + d[6] + d[7];
        tot[q] = __shfl(tl, 0) + __shfl(tl, 16);   // rows 0-7 + rows 8-15
    }

    if (lane == 0) {
        const float cls_loss = tot[0] / fmaxf(tot[1], 1.0f);
        const float l2d      = tot[3] / fmaxf(tot[2], 1.0f);
        const float l3d      = tot[4] / fmaxf(tot[2], 1.0f);
        out[0] = cls_loss + l2d + l3d;
    }
}

// ---------------------------------------------------------------------------
extern "C" void kernel_launch(void* const* d_in, const int* in_sizes, int n_in,
                              void* d_out, int out_size, void* d_ws, size_t ws_size,
                              hipStream_t stream) {
    const float* cls     = (const float*)d_in[0];
    const float* bbox2d  = (const float*)d_in[1];
    const float* bbox3d  = (const float*)d_in[2];
    const float* anchors = (const float*)d_in[3];
    const float* means   = (const float*)d_in[4];
    const float* stds    = (const float*)d_in[5];
    const float* gt2d    = (const float*)d_in[6];
    const float* gt3d    = (const float*)d_in[7];
    const int*   glab    = (const int*)d_in[8];

    int*   g_best   = (int*)d_ws;                                  // 128 ints
    float* partials = (float*)((int*)d_ws + BATCH * MAX_GTS);      // 1980*5 floats

    hipLaunchKernelGGL(k_init, dim3(1), dim3(128), 0, stream, g_best);

    dim3 grid(BLOCKS_PER_B, BATCH);
    hipLaunchKernelGGL(k_best, grid, dim3(BLK), 0, stream,
                       anchors, gt2d, glab, g_best);
    hipLaunchKernelGGL(k_main, grid, dim3(BLK), 0, stream,
                       (const float4*)cls, (const float4*)bbox2d, bbox3d,
                       anchors, means, stds, gt2d, gt3d, glab, g_best, partials);
    hipLaunchKernelGGL(k_final, dim3(1), dim3(32), 0, stream,
                       partials, (float*)d_out);
}